// InteractionPPBlock_85779086836105
// MI455X (gfx1250) — compile-verified
//
#include <hip/hip_runtime.h>
#include <math.h>

typedef __attribute__((ext_vector_type(2))) float v2f;
typedef __attribute__((ext_vector_type(4))) float v4f;
typedef __attribute__((ext_vector_type(8))) float v8f;

#define HID 128
#define INTDIM 64
#define BAS 8
#define NR 6
#define NSR 42
#define E_EDGES 200000
#define T_TRIP 2000000

// swish via single v_exp_f32 + v_rcp_f32 (no IEEE fdiv expansion)
__device__ __forceinline__ float swishf(float v) {
    float e = __builtin_amdgcn_exp2f(-v * 1.44269504088896340736f); // exp(-v)
    return v * __builtin_amdgcn_rcpf(1.0f + e);
}

// One 16x16 output tile: C = A_lds[m0:m0+16, 0:K] @ W[0:K, n0:n0+16]
// A: LDS, row-major with stride lda (padded). W: global, row-major [K, ldw].
// f32 WMMA 16x16x4: A frag 2 VGPRs (lanes 0-15: K=k,k+1 ; lanes 16-31: K=k+2,k+3),
// B frag 2 VGPRs (VGPR0: rows k / k+2 across lane halves, VGPR1: k+1 / k+3).
__device__ __forceinline__ v8f wmma_gemm16(const float* A, int lda, int m0,
                                           const float* W, int ldw, int n0, int K) {
    const int lane = threadIdx.x & 31;
    const int row  = lane & 15;
    const int kh   = (lane >> 4) << 1;          // 0 or 2
    v8f acc = {0.f, 0.f, 0.f, 0.f, 0.f, 0.f, 0.f, 0.f};
    const float* Arow = A + (m0 + row) * lda;
    const float* Wn   = W + n0 + row;           // N position == lane&15
    #pragma unroll 4
    for (int k = 0; k < K; k += 4) {
        v2f a = *(const v2f*)(Arow + k + kh);
        v2f b;
        b.x = Wn[(size_t)(k + kh) * ldw];
        b.y = Wn[(size_t)(k + kh + 1) * ldw];
        acc = __builtin_amdgcn_wmma_f32_16x16x4_f32(false, a, false, b,
                                                    (short)0, acc, false, false);
    }
    return acc;
}

// ---------------- tiny weight fusion: Wrbf[6][128] = w_rbf1[6][8] @ w_rbf2[8][128]
__global__ void k_fuse_rbf(const float* __restrict__ w1, const float* __restrict__ w2,
                           float* __restrict__ Wrbf) {
    int i = blockIdx.x * blockDim.x + threadIdx.x;
    if (i >= NR * HID) return;
    int r = i / HID, n = i % HID;
    float s = 0.f;
    for (int b = 0; b < BAS; ++b) s += w1[r * BAS + b] * w2[b * HID + n];
    Wrbf[i] = s;
}

__global__ void k_zero(float* __restrict__ p, long n) {
    long i = (long)blockIdx.x * blockDim.x + threadIdx.x;
    if (i < n) p[i] = 0.f;
}

// ---------------- edge phase 1
__global__ __launch_bounds__(256)
void k_edge_pre(const float* __restrict__ x, const float* __restrict__ rbf,
                const float* __restrict__ Wrbf,
                const float* __restrict__ w_kj, const float* __restrict__ b_kj,
                const float* __restrict__ w_ji, const float* __restrict__ b_ji,
                const float* __restrict__ w_down,
                float* __restrict__ x_ji_out, float* __restrict__ x_kj_down) {
    __shared__ float lds_x[32 * 132];
    __shared__ float lds_kj[32 * 132];
    __shared__ float lds_rbf[32 * 8];
    __shared__ float lds_wrbf[NR * HID];
    const int tid = threadIdx.x;
    const int wave = tid >> 5;
    const int lane = tid & 31;
    const int e0 = blockIdx.x * 32;

    for (int idx = tid; idx < 32 * 32; idx += 256) {
        int r = idx >> 5, c4 = idx & 31;
        v4f v = *(const v4f*)(x + (size_t)(e0 + r) * HID + c4 * 4);
        *(v4f*)(&lds_x[r * 132 + c4 * 4]) = v;
    }
    for (int idx = tid; idx < 32 * NR; idx += 256) {
        int r = idx / NR, c = idx % NR;
        lds_rbf[r * 8 + c] = rbf[(size_t)(e0 + r) * NR + c];
    }
    for (int idx = tid; idx < NR * HID; idx += 256) lds_wrbf[idx] = Wrbf[idx];
    __syncthreads();

    // x_ji tasks (no per-element branching: separate loops for ji / kj)
    for (int task = wave; task < 16; task += 8) {
        const int m0 = (task >> 3) * 16, n0 = (task & 7) * 16;
        v8f acc = wmma_gemm16(lds_x, 132, m0, w_ji, HID, n0, HID);
        const int n = n0 + (lane & 15);
        const float bias = b_ji[n];
        const int mbase = m0 + ((lane >> 4) << 3);
        #pragma unroll
        for (int i = 0; i < 8; ++i) {
            int m = mbase + i;
            // single-use stream, re-read once by edge_post; keep it out of L2
            __builtin_nontemporal_store(swishf(acc[i] + bias),
                                        &x_ji_out[(size_t)(e0 + m) * HID + n]);
        }
    }
    // x_kj tasks -> LDS (then down-projection)
    for (int task = wave; task < 16; task += 8) {
        const int m0 = (task >> 3) * 16, n0 = (task & 7) * 16;
        v8f acc = wmma_gemm16(lds_x, 132, m0, w_kj, HID, n0, HID);
        const int n = n0 + (lane & 15);
        const float bias = b_kj[n];
        const int mbase = m0 + ((lane >> 4) << 3);
        #pragma unroll
        for (int i = 0; i < 8; ++i) {
            int m = mbase + i;
            float v = swishf(acc[i] + bias);
            float re = 0.f;
            #pragma unroll
            for (int r = 0; r < NR; ++r) re += lds_rbf[m * 8 + r] * lds_wrbf[r * HID + n];
            lds_kj[m * 132 + n] = v * re;
        }
    }
    __syncthreads();

    // down projection: K=128 -> N=64, 8 tasks, one per wave
    {
        const int m0 = (wave >> 2) * 16, n0 = (wave & 3) * 16;
        v8f acc = wmma_gemm16(lds_kj, 132, m0, w_down, INTDIM, n0, HID);
        const int n = n0 + (lane & 15);
        const int mbase = m0 + ((lane >> 4) << 3);
        #pragma unroll
        for (int i = 0; i < 8; ++i)
            x_kj_down[(size_t)(e0 + mbase + i) * INTDIM + n] = swishf(acc[i]);
    }
}

// ---------------- triplet phase
__global__ __launch_bounds__(256)
void k_triplet(const float* __restrict__ sbf, const int* __restrict__ idx_kj,
               const int* __restrict__ idx_ji,
               const float* __restrict__ w_sbf1, const float* __restrict__ w_sbf2,
               const float* __restrict__ x_kj_down, float* __restrict__ agg) {
    __shared__ float lw1[NSR * BAS];
    __shared__ float lw2[BAS * INTDIM];
    __shared__ float lsbf[4][44];
    __shared__ float lt1[4][8];
    const int tid = threadIdx.x;
    for (int i = tid; i < NSR * BAS; i += 256) lw1[i] = w_sbf1[i];
    for (int i = tid; i < BAS * INTDIM; i += 256) lw2[i] = w_sbf2[i];
    const int ti = tid >> 6;        // triplet slot 0..3
    const int l  = tid & 63;        // channel / helper lane
    const long t = (long)blockIdx.x * 4 + ti;
    if (l < NSR) lsbf[ti][l] = __builtin_nontemporal_load(&sbf[t * NSR + l]); // 336MB stream
    __syncthreads();
    if (l < BAS) {
        float s = 0.f;
        #pragma unroll
        for (int i = 0; i < NSR; ++i) s += lsbf[ti][i] * lw1[i * BAS + l];
        lt1[ti][l] = s;
    }
    __syncthreads();
    float s = 0.f;
    #pragma unroll
    for (int j = 0; j < BAS; ++j) s += lt1[ti][j] * lw2[j * INTDIM + l];
    const int ekj = idx_kj[t];
    const int eji = idx_ji[t];
    const float m = x_kj_down[(size_t)ekj * INTDIM + l] * s;   // 51MB table, L2-resident
    atomicAdd(&agg[(size_t)eji * INTDIM + l], m);              // 51MB target, L2-resident
}

// ---------------- residual block on a 32x128 LDS tile
__device__ __forceinline__ void res_block(float* lds_h, float* lds_t,
                                          const float* w1, const float* b1,
                                          const float* w2, const float* b2,
                                          int wave, int lane) {
    for (int task = wave; task < 16; task += 8) {
        const int m0 = (task >> 3) * 16, n0 = (task & 7) * 16;
        v8f acc = wmma_gemm16(lds_h, 132, m0, w1, HID, n0, HID);
        const int n = n0 + (lane & 15);
        const float bias = b1[n];
        const int mbase = m0 + ((lane >> 4) << 3);
        #pragma unroll
        for (int i = 0; i < 8; ++i)
            lds_t[(mbase + i) * 132 + n] = swishf(acc[i] + bias);
    }
    __syncthreads();
    for (int task = wave; task < 16; task += 8) {
        const int m0 = (task >> 3) * 16, n0 = (task & 7) * 16;
        v8f acc = wmma_gemm16(lds_t, 132, m0, w2, HID, n0, HID);
        const int n = n0 + (lane & 15);
        const float bias = b2[n];
        const int mbase = m0 + ((lane >> 4) << 3);
        #pragma unroll
        for (int i = 0; i < 8; ++i)
            lds_h[(mbase + i) * 132 + n] += swishf(acc[i] + bias);
    }
    __syncthreads();
}

// ---------------- edge phase 2
__global__ __launch_bounds__(256)
void k_edge_post(const float* __restrict__ agg, const float* __restrict__ x_ji,
                 const float* __restrict__ x, const float* __restrict__ w_up,
                 const float* __restrict__ rb0_w1, const float* __restrict__ rb0_b1,
                 const float* __restrict__ rb0_w2, const float* __restrict__ rb0_b2,
                 const float* __restrict__ w_lin, const float* __restrict__ b_lin,
                 const float* __restrict__ ra0_w1, const float* __restrict__ ra0_b1,
                 const float* __restrict__ ra0_w2, const float* __restrict__ ra0_b2,
                 const float* __restrict__ ra1_w1, const float* __restrict__ ra1_b1,
                 const float* __restrict__ ra1_w2, const float* __restrict__ ra1_b2,
                 float* __restrict__ out) {
    __shared__ float lds_h[32 * 132];
    __shared__ float lds_t[32 * 132];
    __shared__ float lds_a[32 * 68];
    const int tid = threadIdx.x, wave = tid >> 5, lane = tid & 31;
    const int e0 = blockIdx.x * 32;

    for (int idx = tid; idx < 32 * 16; idx += 256) {
        int r = idx >> 4, c4 = idx & 15;
        v4f v = *(const v4f*)(agg + (size_t)(e0 + r) * INTDIM + c4 * 4);
        *(v4f*)(&lds_a[r * 68 + c4 * 4]) = v;
    }
    __syncthreads();

    // h = swish(agg @ w_up) + x_ji   (K=64)
    for (int task = wave; task < 16; task += 8) {
        const int m0 = (task >> 3) * 16, n0 = (task & 7) * 16;
        v8f acc = wmma_gemm16(lds_a, 68, m0, w_up, HID, n0, INTDIM);
        const int n = n0 + (lane & 15);
        const int mbase = m0 + ((lane >> 4) << 3);
        #pragma unroll
        for (int i = 0; i < 8; ++i) {
            int m = mbase + i;
            lds_h[m * 132 + n] = swishf(acc[i]) + x_ji[(size_t)(e0 + m) * HID + n];
        }
    }
    __syncthreads();

    res_block(lds_h, lds_t, rb0_w1, rb0_b1, rb0_w2, rb0_b2, wave, lane);

    // h = swish(h @ w_lin + b_lin) + x   (in-place on lds_h -> two-phase)
    {
        v8f acc[2];
        int s2 = 0;
        for (int task = wave; task < 16; task += 8, ++s2)
            acc[s2] = wmma_gemm16(lds_h, 132, (task >> 3) * 16, w_lin, HID, (task & 7) * 16, HID);
        __syncthreads();
        s2 = 0;
        for (int task = wave; task < 16; task += 8, ++s2) {
            const int m0 = (task >> 3) * 16, n0 = (task & 7) * 16;
            const int n = n0 + (lane & 15);
            const float bias = b_lin[n];
            const int mbase = m0 + ((lane >> 4) << 3);
            #pragma unroll
            for (int i = 0; i < 8; ++i) {
                int m = mbase + i;
                lds_h[m * 132 + n] = swishf(acc[s2][i] + bias) + x[(size_t)(e0 + m) * HID + n];
            }
        }
        __syncthreads();
    }

    res_block(lds_h, lds_t, ra0_w1, ra0_b1, ra0_w2, ra0_b2, wave, lane);
    res_block(lds_h, lds_t, ra1_w1, ra1_b1, ra1_w2, ra1_b2, wave, lane);

    for (int idx = tid; idx < 32 * 32; idx += 256) {
        int r = idx >> 5, c4 = idx & 31;
        v4f v = *(const v4f*)(&lds_h[r * 132 + c4 * 4]);
        __builtin_nontemporal_store(v, (v4f*)(out + (size_t)(e0 + r) * HID + c4 * 4));
    }
}

extern "C" void kernel_launch(void* const* d_in, const int* in_sizes, int n_in,
                              void* d_out, int out_size, void* d_ws, size_t ws_size,
                              hipStream_t stream) {
    const float* x      = (const float*)d_in[0];
    const float* rbf    = (const float*)d_in[1];
    const float* sbf    = (const float*)d_in[2];
    const int*   idx_kj = (const int*)d_in[3];
    const int*   idx_ji = (const int*)d_in[4];
    const float* w_rbf1 = (const float*)d_in[5];
    const float* w_rbf2 = (const float*)d_in[6];
    const float* w_sbf1 = (const float*)d_in[7];
    const float* w_sbf2 = (const float*)d_in[8];
    const float* w_kj   = (const float*)d_in[9];
    const float* b_kj   = (const float*)d_in[10];
    const float* w_ji   = (const float*)d_in[11];
    const float* b_ji   = (const float*)d_in[12];
    const float* w_down = (const float*)d_in[13];
    const float* w_up   = (const float*)d_in[14];
    const float* rb0_w1 = (const float*)d_in[15];
    const float* rb0_b1 = (const float*)d_in[16];
    const float* rb0_w2 = (const float*)d_in[17];
    const float* rb0_b2 = (const float*)d_in[18];
    const float* w_lin  = (const float*)d_in[19];
    const float* b_lin  = (const float*)d_in[20];
    const float* ra0_w1 = (const float*)d_in[21];
    const float* ra0_b1 = (const float*)d_in[22];
    const float* ra0_w2 = (const float*)d_in[23];
    const float* ra0_b2 = (const float*)d_in[24];
    const float* ra1_w1 = (const float*)d_in[25];
    const float* ra1_b1 = (const float*)d_in[26];
    const float* ra1_w2 = (const float*)d_in[27];
    const float* ra1_b2 = (const float*)d_in[28];

    char* ws = (char*)d_ws;
    const size_t off_xji  = 0;
    const size_t off_xkjd = off_xji  + (size_t)E_EDGES * HID * 4;
    const size_t off_agg  = off_xkjd + (size_t)E_EDGES * INTDIM * 4;
    const size_t off_wrbf = off_agg  + (size_t)E_EDGES * INTDIM * 4;
    float* x_ji      = (float*)(ws + off_xji);
    float* x_kj_down = (float*)(ws + off_xkjd);
    float* agg       = (float*)(ws + off_agg);
    float* Wrbf      = (float*)(ws + off_wrbf);

    k_fuse_rbf<<<(NR * HID + 255) / 256, 256, 0, stream>>>(w_rbf1, w_rbf2, Wrbf);

    const long aggN = (long)E_EDGES * INTDIM;
    k_zero<<<(int)((aggN + 255) / 256), 256, 0, stream>>>(agg, aggN);

    k_edge_pre<<<E_EDGES / 32, 256, 0, stream>>>(x, rbf, Wrbf, w_kj, b_kj, w_ji, b_ji,
                                                 w_down, x_ji, x_kj_down);

    k_triplet<<<T_TRIP / 4, 256, 0, stream>>>(sbf, idx_kj, idx_ji, w_sbf1, w_sbf2,
                                              x_kj_down, agg);

    k_edge_post<<<E_EDGES / 32, 256, 0, stream>>>(agg, x_ji, x, w_up,
                                                  rb0_w1, rb0_b1, rb0_w2, rb0_b2,
                                                  w_lin, b_lin,
                                                  ra0_w1, ra0_b1, ra0_w2, ra0_b2,
                                                  ra1_w1, ra1_b1, ra1_w2, ra1_b2,
                                                  (float*)d_out);
}